// ACmix_42752104464784
// MI455X (gfx1250) — compile-verified
//
#include <hip/hip_runtime.h>

typedef __attribute__((ext_vector_type(2))) float v2f;
typedef __attribute__((ext_vector_type(8))) float v8f;

#define NHEADS 4
#define K2     9
#define CCH    256      // C
#define LPIX   4096     // H*W
#define QKVCH  768      // 3C
#define BATCH  8

// -----------------------------------------------------------------------------
// Generic fp32 WMMA GEMM:  D[b] = W (MxK, row major)  x  S[b] (KxN, row major)
// One wave (32 threads) per 32(M) x 64(N) output tile, K stepped by 4 via
// v_wmma_f32_16x16x4_f32.  8 accumulators (2 M-subtiles x 4 N-subtiles).
// -----------------------------------------------------------------------------
__global__ __launch_bounds__(32) void wmma_gemm_f32(
    const float* __restrict__ W,   // M x K
    const float* __restrict__ S,   // per-batch K x N
    float*       __restrict__ D,   // per-batch M x N
    int K, int N,
    long sBatch, long dBatch)
{
  const int lane = threadIdx.x & 31;
  const int half = lane >> 4;     // 0 or 1 (lane group)
  const int lm   = lane & 15;
  const int n0   = blockIdx.x * 64;
  const int m0   = blockIdx.y * 32;
  const int b    = blockIdx.z;

  const float* __restrict__ Sb = S + (long)b * sBatch;
  float*       __restrict__ Db = D + (long)b * dBatch;

  const float* w0 = W + (long)(m0 + lm)      * K;   // A rows for subtile 0
  const float* w1 = W + (long)(m0 + 16 + lm) * K;   // A rows for subtile 1

  v8f acc[2][4] = {};

  for (int k0 = 0; k0 < K; k0 += 4) {
    const int kA = k0 + 2 * half;          // A/B layout: lanes 16-31 hold K+2
    v2f a0, a1;
    a0.x = w0[kA];     a0.y = w0[kA + 1];
    a1.x = w1[kA];     a1.y = w1[kA + 1];

    const float* s0 = Sb + (long)kA * N + n0 + lm;  // row kA
    const float* s1 = s0 + N;                       // row kA+1
#pragma unroll
    for (int j = 0; j < 4; ++j) {
      v2f bb;
      bb.x = s0[j * 16];
      bb.y = s1[j * 16];
      acc[0][j] = __builtin_amdgcn_wmma_f32_16x16x4_f32(
          false, a0, false, bb, (short)0, acc[0][j], false, false);
      acc[1][j] = __builtin_amdgcn_wmma_f32_16x16x4_f32(
          false, a1, false, bb, (short)0, acc[1][j], false, false);
    }
  }

#pragma unroll
  for (int mi = 0; mi < 2; ++mi) {
#pragma unroll
    for (int j = 0; j < 4; ++j) {
#pragma unroll
      for (int r = 0; r < 8; ++r) {
        const int m = m0 + mi * 16 + r + 8 * half;   // C/D layout
        Db[(long)m * N + n0 + j * 16 + lm] = acc[mi][j][r];
      }
    }
  }
}

// -----------------------------------------------------------------------------
// Fused mid-stage. One thread per (b, head, pixel).
//   logits[k] = q . k_patch[k]           (9 x 64-dot, zero-padded window)
//   attn      = softmax(logits)
//   ker[k]    = w_kg[h,k,:] . qkv_group + b_kg   (9 x 192-dot)
//   tmp[d]    = sum_k (alpha*ker[k] + beta*attn[k]) * v_patch[d,k]
// Adjacent threads = adjacent pixels -> coalesced along L.
// -----------------------------------------------------------------------------
__global__ __launch_bounds__(256) void acmix_mid(
    const float* __restrict__ qkv,    // (B, 768, 4096)
    const float* __restrict__ w_kg,   // (36, 192)
    const float* __restrict__ b_kg,   // (36)
    const float* __restrict__ alphaP,
    const float* __restrict__ betaP,
    float*       __restrict__ tmp)    // (B, 256, 4096)
{
  const int gid = blockIdx.x * blockDim.x + threadIdx.x;
  const int l   = gid & (LPIX - 1);
  const int hh  = (gid >> 12) & (NHEADS - 1);
  const int b   = gid >> 14;
  const int y   = l >> 6;
  const int x   = l & 63;

  const float alpha = alphaP[0];
  const float beta  = betaP[0];
  const float* __restrict__ base = qkv + (long)b * QKVCH * LPIX;

  // 3x3 window offsets with zero padding (pad elements contribute 0).
  int   ll[K2];
  float inb[K2];
#pragma unroll
  for (int kk = 0; kk < K2; ++kk) {
    const int yy = y + kk / 3 - 1;
    const int xx = x + kk % 3 - 1;
    const bool ok = (yy >= 0) && (yy < 64) && (xx >= 0) && (xx < 64);
    inb[kk] = ok ? 1.0f : 0.0f;
    ll[kk]  = ok ? (yy * 64 + xx) : l;   // clamped addr, masked by inb
  }

  // Cache this head's q vector (64 ch).
  float q[64];
  const float* qb = base + (long)(hh * 64) * LPIX + l;
#pragma unroll
  for (int dd = 0; dd < 64; ++dd) q[dd] = qb[(long)dd * LPIX];

  // Attention logits over the 9 taps.
  float logits[K2];
  const float* kb = base + (long)(CCH + hh * 64) * LPIX;
#pragma unroll
  for (int kk = 0; kk < K2; ++kk) {
    const float* kp = kb + ll[kk];
    float s = 0.0f;
#pragma unroll 8
    for (int dd = 0; dd < 64; ++dd) s += q[dd] * kp[(long)dd * LPIX];
    logits[kk] = s * inb[kk];            // OOB patch is all-zero -> logit 0
  }

  // Softmax over k2 (OOB taps included, matching the reference).
  float mx = logits[0];
#pragma unroll
  for (int kk = 1; kk < K2; ++kk) mx = fmaxf(mx, logits[kk]);
  float att[K2], den = 0.0f;
#pragma unroll
  for (int kk = 0; kk < K2; ++kk) { att[kk] = __expf(logits[kk] - mx); den += att[kk]; }
  const float rden = 1.0f / den;

  // Dynamic kernel from grouped 1x1 conv on cat(q,k,v): group hh = ch [hh*192, hh*192+192).
  float ker[K2];
#pragma unroll
  for (int kk = 0; kk < K2; ++kk) ker[kk] = b_kg[hh * K2 + kk];
  const float* cb = base + (long)(hh * 192) * LPIX + l;
  for (int c = 0; c < 192; ++c) {
    const float cv = cb[(long)c * LPIX];
#pragma unroll
    for (int kk = 0; kk < K2; ++kk)
      ker[kk] += w_kg[(hh * K2 + kk) * 192 + c] * cv;   // scalar (uniform) loads
  }

  // Combined tap weights: alpha*conv + beta*attn collapses into one 9-tap pass.
  float wgt[K2];
#pragma unroll
  for (int kk = 0; kk < K2; ++kk)
    wgt[kk] = (alpha * ker[kk] + beta * att[kk] * rden) * inb[kk];

  // Apply to v patches and write the mixed result.
  const float* vb = base + (long)(2 * CCH + hh * 64) * LPIX;
  float* ob = tmp + (long)b * CCH * LPIX + (long)(hh * 64) * LPIX + l;
#pragma unroll 4
  for (int dd = 0; dd < 64; ++dd) {
    const float* vp = vb + (long)dd * LPIX;
    float o = 0.0f;
#pragma unroll
    for (int kk = 0; kk < K2; ++kk) o += wgt[kk] * vp[ll[kk]];
    ob[(long)dd * LPIX] = o;
  }
}

// -----------------------------------------------------------------------------
// Launch: qkv GEMM (WMMA) -> fused mid stage -> projection GEMM (WMMA)
// -----------------------------------------------------------------------------
extern "C" void kernel_launch(void* const* d_in, const int* in_sizes, int n_in,
                              void* d_out, int out_size, void* d_ws, size_t ws_size,
                              hipStream_t stream) {
  const float* x      = (const float*)d_in[0];
  const float* w_qkv  = (const float*)d_in[1];
  const float* w_kg   = (const float*)d_in[2];
  const float* b_kg   = (const float*)d_in[3];
  const float* alpha  = (const float*)d_in[4];
  const float* beta   = (const float*)d_in[5];
  const float* w_proj = (const float*)d_in[6];
  float* out = (float*)d_out;

  float* qkv = (float*)d_ws;                              // B*768*4096 f32 (96 MiB)
  float* tmp = qkv + (size_t)BATCH * QKVCH * LPIX;        // B*256*4096 f32 (32 MiB)

  // 1) qkv = w_qkv @ x   (M=768, K=256, N=4096 per batch)
  dim3 g1(LPIX / 64, QKVCH / 32, BATCH);
  wmma_gemm_f32<<<g1, 32, 0, stream>>>(w_qkv, x, qkv, CCH, LPIX,
                                       (long)CCH * LPIX, (long)QKVCH * LPIX);

  // 2) fused attention + dynamic-conv mixing
  const int nthreads = BATCH * NHEADS * LPIX;             // 131072
  acmix_mid<<<nthreads / 256, 256, 0, stream>>>(qkv, w_kg, b_kg, alpha, beta, tmp);

  // 3) out = w_proj @ tmp  (M=256, K=256, N=4096 per batch)
  dim3 g3(LPIX / 64, CCH / 32, BATCH);
  wmma_gemm_f32<<<g3, 32, 0, stream>>>(w_proj, tmp, out, CCH, LPIX,
                                       (long)CCH * LPIX, (long)CCH * LPIX);
}